// AttentionModel_36764920054034
// MI455X (gfx1250) — compile-verified
//
#include <hip/hip_runtime.h>
#include <math.h>

// Problem constants (match reference)
#define BB   64
#define SS   100
#define EE   512
#define HH   1024      // 2*E
#define VTT  32000
#define XK   (EE + HH) // 1536

typedef __attribute__((ext_vector_type(2))) float v2f;
typedef __attribute__((ext_vector_type(4))) float v4f;
typedef __attribute__((ext_vector_type(8))) float v8f;

// ---------------------------------------------------------------------------
// K1: enc[b,s,:] = concat(emb_in[src[b,s]], emb_pos[pos[b,s]])   [B,S,H]
// one block per (b,s); threads 0..127 copy words (E/4 float4s), 128..255 pos
// ---------------------------------------------------------------------------
__global__ void gather_enc(const int* __restrict__ src, const int* __restrict__ pos,
                           const float* __restrict__ emb_in, const float* __restrict__ emb_pos,
                           float* __restrict__ enc) {
    const int bs = blockIdx.x;                 // b*S + s
    const int t  = threadIdx.x;
    float* dst = enc + (size_t)bs * HH;
    if (t < 128) {
        const int row = src[bs];
        const v4f* srcv = (const v4f*)(emb_in + (size_t)row * EE);
        ((v4f*)dst)[t] = srcv[t];              // E/4 = 128 float4s
    } else {
        const int t2 = t - 128;
        const int row = pos[bs];
        const v4f* srcv = (const v4f*)(emb_pos + (size_t)row * EE);
        ((v4f*)(dst + EE))[t2] = srcv[t2];
    }
}

// ---------------------------------------------------------------------------
// K2: mean_enc[b,h] = (1/S) * sum_s enc[b,s,h]
// ---------------------------------------------------------------------------
__global__ void mean_over_s(const float* __restrict__ enc, float* __restrict__ mean_enc) {
    const int h = blockIdx.x * 256 + threadIdx.x;
    const int b = blockIdx.y;
    const float* p = enc + (size_t)b * SS * HH + h;
    float acc = 0.f;
    #pragma unroll 4
    for (int s = 0; s < SS; ++s) acc += p[(size_t)s * HH];
    mean_enc[(size_t)b * HH + h] = acc * (1.0f / SS);
}

// ---------------------------------------------------------------------------
// WMMA GEMM: D[M,N] = A[M,K] @ W[N,K]^T + bias[N] (+ Cin) (tanh optional)
// One wave owns a 16-wide N stripe and MT m-tiles (W frag reused MT times).
// b128 per-lane loads: lane<16 gets floats k8..k8+3, lane>=16 gets k8+4..k8+7.
// WMMA slot mapping j -> {k8,k8+1,k8+4,k8+5} (frag1) / {k8+2,k8+3,k8+6,k8+7}
// (frag2) is identical for A and B, so the K contraction is exact.
// C/D layout: VGPR v, lane l -> M = (l>>4)*8 + v, N = l&15.
// ---------------------------------------------------------------------------
template <int MT, bool ACC_IN, bool TANH, bool NT>
__global__ void wmma_gemm_bias(const float* __restrict__ A, const float* __restrict__ W,
                               const float* __restrict__ bias, const float* __restrict__ Cin,
                               float* __restrict__ D, int N, int K) {
    const int lane  = threadIdx.x & 31;
    const int wave  = threadIdx.x >> 5;
    const int nTile = blockIdx.x * 4 + wave;   // 4 waves per block
    const int m0 = blockIdx.y * (16 * MT);
    const int n0 = nTile * 16;

    const int koff = (lane >> 4) << 2;         // 0 or 4: K sub-offset per half-wave
    const float* Arow = A + (size_t)(m0 + (lane & 15)) * K + koff;
    const float* Wrow = W + (size_t)(n0 + (lane & 15)) * K + koff;

    v8f acc[MT];
    #pragma unroll
    for (int mt = 0; mt < MT; ++mt) acc[mt] = (v8f){};

    for (int kk = 0; kk < K; kk += 64) {
        if (!NT) {
            __builtin_prefetch(Wrow + kk + 64, 0, 1);   // global_prefetch_b8
            __builtin_prefetch(Arow + kk + 64, 0, 1);
        }
        #pragma unroll
        for (int k8 = 0; k8 < 64; k8 += 8) {
            v4f w = NT ? __builtin_nontemporal_load((const v4f*)(Wrow + kk + k8))
                       : *(const v4f*)(Wrow + kk + k8);
            v2f b1 = {w[0], w[1]};
            v2f b2 = {w[2], w[3]};
            #pragma unroll
            for (int mt = 0; mt < MT; ++mt) {
                v4f a = *(const v4f*)(Arow + (size_t)mt * 16 * K + kk + k8);
                v2f a1 = {a[0], a[1]};
                v2f a2 = {a[2], a[3]};
                // 8 args: (neg_a, A, neg_b, B, c_mod, C, reuse_a, reuse_b)
                acc[mt] = __builtin_amdgcn_wmma_f32_16x16x4_f32(
                    false, a1, false, b1, (short)0, acc[mt], false, false);
                acc[mt] = __builtin_amdgcn_wmma_f32_16x16x4_f32(
                    false, a2, false, b2, (short)0, acc[mt], false, false);
            }
        }
    }

    const float bn = bias[n0 + (lane & 15)];
    const int nc   = n0 + (lane & 15);
    #pragma unroll
    for (int mt = 0; mt < MT; ++mt) {
        const int mBase = m0 + mt * 16 + ((lane >> 4) << 3);
        #pragma unroll
        for (int v = 0; v < 8; ++v) {
            float r = acc[mt][v] + bn;
            if (ACC_IN) r += Cin[(size_t)(mBase + v) * N + nc];
            if (TANH)   r = tanhf(r);
            D[(size_t)(mBase + v) * N + nc] = r;
        }
    }
}

// ---------------------------------------------------------------------------
// K4: attention per batch: energy -> softmax over S -> context = mean_s a*enc
// one block per b, 256 threads (8 wave32s)
// ---------------------------------------------------------------------------
__global__ void attention_kernel(const float* __restrict__ enc, const float* __restrict__ h0,
                                 float* __restrict__ context) {
    __shared__ float h0s[HH];
    __shared__ float e[128];
    __shared__ float s_inv;

    const int b    = blockIdx.x;
    const int t    = threadIdx.x;
    const int lane = t & 31;
    const int wave = t >> 5;
    const float* encB = enc + (size_t)b * SS * HH;

    #pragma unroll
    for (int i = t; i < HH; i += 256) h0s[i] = h0[(size_t)b * HH + i];
    __syncthreads();

    // phase A: energy[s] = dot(h0, enc[b,s,:])
    for (int s = wave; s < SS; s += 8) {
        const float* row = encB + (size_t)s * HH;
        float acc = 0.f;
        #pragma unroll 4
        for (int j = lane; j < HH; j += 32) acc += h0s[j] * row[j];
        #pragma unroll
        for (int m = 16; m >= 1; m >>= 1) acc += __shfl_xor(acc, m, 32);
        if (lane == 0) e[s] = acc;
    }
    __syncthreads();

    // phase B: softmax over S=100 (wave 0), store exp() in e[], 1/(sum*S) in s_inv
    if (wave == 0) {
        float mx = -3.4e38f;
        for (int i = lane; i < 128; i += 32) {
            float v = (i < SS) ? e[i] : -3.4e38f;
            mx = fmaxf(mx, v);
        }
        #pragma unroll
        for (int m = 16; m >= 1; m >>= 1) mx = fmaxf(mx, __shfl_xor(mx, m, 32));
        float sum = 0.f;
        for (int i = lane; i < 128; i += 32) {
            float ex = (i < SS) ? __expf(e[i] - mx) : 0.f;
            e[i] = ex;
            sum += ex;
        }
        #pragma unroll
        for (int m = 16; m >= 1; m >>= 1) sum += __shfl_xor(sum, m, 32);
        if (lane == 0) s_inv = 1.0f / (sum * (float)SS);
    }
    __syncthreads();

    // phase C: context[b,h] = s_inv * sum_s e[s]*enc[b,s,h]
    const float inv = s_inv;
    #pragma unroll
    for (int c = 0; c < 4; ++c) {
        const int h = t + c * 256;
        float acc = 0.f;
        #pragma unroll 4
        for (int s = 0; s < SS; ++s) acc += e[s] * encB[(size_t)s * HH + h];
        context[(size_t)b * HH + h] = acc * inv;
    }
}

// ---------------------------------------------------------------------------
// K5: x[b,:] = concat(emb_out[tw[b]] (E), context[b] (H))
// ---------------------------------------------------------------------------
__global__ void build_x(const int* __restrict__ tw, const float* __restrict__ emb_out,
                        const float* __restrict__ context, float* __restrict__ x) {
    const int b = blockIdx.x;
    const int t = threadIdx.x;
    float* xb = x + (size_t)b * XK;
    if (t < 128) {
        const int row = tw[b];
        ((v4f*)xb)[t] = ((const v4f*)(emb_out + (size_t)row * EE))[t];
    } else {
        const int t2 = t - 128;
        ((v4f*)(xb + EE))[t2] = ((const v4f*)(context + (size_t)b * HH))[t2];
    }
}

// ---------------------------------------------------------------------------
extern "C" void kernel_launch(void* const* d_in, const int* in_sizes, int n_in,
                              void* d_out, int out_size, void* d_ws, size_t ws_size,
                              hipStream_t stream) {
    (void)in_sizes; (void)n_in; (void)out_size; (void)ws_size;

    const int*   src      = (const int*)  d_in[0];
    const int*   pos      = (const int*)  d_in[1];
    const int*   tw       = (const int*)  d_in[2];
    // d_in[3] source_lengths, d_in[4] max_len: unused (all lengths == S)
    const float* emb_in   = (const float*)d_in[5];
    const float* emb_out  = (const float*)d_in[6];
    const float* emb_pos  = (const float*)d_in[7];
    const float* W_scale  = (const float*)d_in[8];
    const float* b_scale  = (const float*)d_in[9];
    const float* W_ih     = (const float*)d_in[10];
    const float* b_ih     = (const float*)d_in[11];
    const float* W_hh     = (const float*)d_in[12];
    const float* b_hh     = (const float*)d_in[13];
    const float* W_proj   = (const float*)d_in[14];
    const float* b_proj   = (const float*)d_in[15];

    float* out   = (float*)d_out;                       // [B, VT]
    float* h_out = out + (size_t)BB * VTT;              // [B, H]

    float* ws       = (float*)d_ws;
    float* enc      = ws;                                // B*S*H
    float* mean_enc = enc      + (size_t)BB * SS * HH;   // B*H
    float* h0       = mean_enc + (size_t)BB * HH;        // B*H
    float* context  = h0       + (size_t)BB * HH;        // B*H
    float* x        = context  + (size_t)BB * HH;        // B*(E+H)
    float* hpre     = x        + (size_t)BB * XK;        // B*H

    // 1. build enc
    gather_enc<<<BB * SS, 256, 0, stream>>>(src, pos, emb_in, emb_pos, enc);
    // 2. mean over S
    mean_over_s<<<dim3(HH / 256, BB), 256, 0, stream>>>(enc, mean_enc);
    // 3. h0 = mean_enc @ W_scale^T + b_scale   (M=64,N=1024,K=1024)
    wmma_gemm_bias<1, false, false, false><<<dim3(HH / 64, BB / 16), 128, 0, stream>>>(
        mean_enc, W_scale, b_scale, nullptr, h0, HH, HH);
    // 4. attention -> context
    attention_kernel<<<BB, 256, 0, stream>>>(enc, h0, context);
    // 5. x = concat(emb_out[tw], context)
    build_x<<<BB, 384, 0, stream>>>(tw, emb_out, context, x);
    // 6a. hpre = x @ W_ih^T + b_ih             (M=64,N=1024,K=1536)
    wmma_gemm_bias<1, false, false, false><<<dim3(HH / 64, BB / 16), 128, 0, stream>>>(
        x, W_ih, b_ih, nullptr, hpre, HH, XK);
    // 6b. h = tanh(hpre + h0 @ W_hh^T + b_hh)  (M=64,N=1024,K=1024)
    wmma_gemm_bias<1, true, true, false><<<dim3(HH / 64, BB / 16), 128, 0, stream>>>(
        h0, W_hh, b_hh, hpre, h_out, HH, HH);
    // 7. out = h @ W_proj^T + b_proj           (M=64,N=32000,K=1024)
    //    MT=4: each wave computes all of M, W_proj streamed exactly once (NT);
    //    A (256 KB) stays L2-resident across the 500 blocks.
    wmma_gemm_bias<4, false, false, true><<<dim3(VTT / 64, 1), 128, 0, stream>>>(
        h_out, W_proj, b_proj, nullptr, out, VTT, HH);
}